// NSM_46712064312161
// MI455X (gfx1250) — compile-verified
//
#include <hip/hip_runtime.h>
#include <stdint.h>

typedef __attribute__((ext_vector_type(2))) float v2f;
typedef __attribute__((ext_vector_type(4))) float v4f;
typedef __attribute__((ext_vector_type(8))) float v8f;

#define OUT_DIM 512
#define IN_DIM  1024
#define BATCH_N 4096

#define BM 64
#define BN 128
#define BK 32
#define LDA_SH (BK + 4)    // A pad: conflict-free b64 frag reads, 16B-aligned stores
#define LDB_P  (BN + 16)   // B pad: half-waves land on disjoint bank halves

// D[M x N] = (bias ? bias : 0) + A[M x K] @ B[K x N]
// Grid: (N/BN, M/BM), 256 threads = 8 waves (4 M-waves x 2 N-waves).
// Wave computes a 16x64 strip = 4 x (16x16) fp32 WMMA tiles.
// Double-buffered LDS; A tile staged with GLOBAL_LOAD_ASYNC_TO_LDS_B128.
__global__ __launch_bounds__(256)
void nsm_wmma_gemm(const float* __restrict__ A, int K,
                   const float* __restrict__ B, int N,
                   const float* __restrict__ bias,
                   float* __restrict__ D)
{
    __shared__ __align__(16) float shA[2][BM][LDA_SH];
    // Pair-interleaved B: shB[b][p][c] = {B[2p][c], B[2p+1][c]} -> one ds_load_b64 per frag
    __shared__ __align__(16) float shB[2][BK / 2][LDB_P][2];

    const int tid  = threadIdx.x;
    const int lane = tid & 31;
    const int wave = tid >> 5;
    const int wm   = wave >> 1;    // 0..3 : 16 rows each
    const int wn   = wave & 1;     // 0..1 : 64 cols each
    const int half = lane >> 4;
    const int l16  = lane & 15;

    const int blockRow = blockIdx.y * BM;
    const int blockCol = blockIdx.x * BN;

    v8f acc[4] = {};
    v4f breg[2][2];

    // ---- A tile (64x32) -> LDS buffer nb via CDNA5 async DMA (ASYNCcnt) ----
    auto loadA_async = [&](int ktn, int nb) {
        #pragma unroll
        for (int it = 0; it < 2; ++it) {
            int f = tid + it * 256;
            int r = f >> 3;                 // 8 float4 per row
            int c = (f & 7) << 2;
            unsigned ldsa = (unsigned)(uintptr_t)&shA[nb][r][c];  // LDS aperture: low 32b = DS addr
            const float* gp = &A[(size_t)(blockRow + r) * K + ktn + c];
            asm volatile("global_load_async_to_lds_b128 %0, %1, off"
                         :: "v"(ldsa), "v"(gp) : "memory");
        }
    };
    // ---- B tile (32x128): global -> regs (early), pair-interleave in VALU ----
    auto loadB = [&](int ktn) {
        #pragma unroll
        for (int it = 0; it < 2; ++it) {
            int f = tid + it * 256;
            int p = f >> 5;                 // row pair 0..15
            int c = (f & 31) << 2;          // col 0..124
            const float* g0 = &B[(size_t)(ktn + 2 * p) * N + blockCol + c];
            v4f b0 = *(const v4f*)g0;
            v4f b1 = *(const v4f*)(g0 + N);
            v4f s0 = {b0.x, b1.x, b0.y, b1.y};
            v4f s1 = {b0.z, b1.z, b0.w, b1.w};
            breg[it][0] = s0;
            breg[it][1] = s1;
        }
    };
    auto storeB = [&](int nb) {            // regs -> LDS (late, after compute)
        #pragma unroll
        for (int it = 0; it < 2; ++it) {
            int f = tid + it * 256;
            int p = f >> 5;
            int c = (f & 31) << 2;
            *(v4f*)&shB[nb][p][c][0]     = breg[it][0];
            *(v4f*)&shB[nb][p][c + 2][0] = breg[it][1];
        }
    };

    const int T = K / BK;
    loadA_async(0, 0);
    loadB(0);
    storeB(0);

    for (int t = 0; t < T; ++t) {
        const int cur = t & 1;
        // async writes into shA[cur] must land before anyone signals the barrier
        asm volatile("s_wait_asynccnt 0x0" ::: "memory");
        __syncthreads();

        if (t + 1 < T) {                   // prefetch next tile while computing
            loadA_async((t + 1) * BK, cur ^ 1);
            loadB((t + 1) * BK);
        }

        #pragma unroll
        for (int kk = 0; kk < BK; kk += 4) {
            // A frag 16x4 f32: lanes 0-15 -> {K=kk,kk+1}, lanes 16-31 -> {kk+2,kk+3}
            v2f a = *(const v2f*)&shA[cur][wm * 16 + l16][kk + 2 * half];
            // B frag 4x16 f32: VGPR0 = rows {kk, kk+2}, VGPR1 = rows {kk+1, kk+3}
            v2f bf[4];
            #pragma unroll
            for (int j = 0; j < 4; ++j) {
                int bc = wn * 64 + j * 16 + l16;
                bf[j] = *(const v2f*)&shB[cur][(kk >> 1) + half][bc][0];
            }
            #pragma unroll
            for (int j = 0; j < 4; ++j)
                acc[j] = __builtin_amdgcn_wmma_f32_16x16x4_f32(
                    false, a, false, bf[j], (short)0, acc[j], false, false);
        }

        if (t + 1 < T) storeB(cur ^ 1);
    }

    // ---- epilogue: VGPR v, lane l holds D[v + 8*half][l16] of each 16x16 tile ----
    const int row0 = blockRow + wm * 16 + half * 8;
    #pragma unroll
    for (int j = 0; j < 4; ++j) {
        int col = blockCol + wn * 64 + j * 16 + l16;
        if (bias != nullptr) {
            #pragma unroll
            for (int v = 0; v < 8; ++v) {
                size_t idx = (size_t)(row0 + v) * N + col;
                D[idx] = bias[idx] + acc[j][v];
            }
        } else {
            #pragma unroll
            for (int v = 0; v < 8; ++v)
                D[(size_t)(row0 + v) * N + col] = acc[j][v];
        }
    }
}

// P = -M and I = identity (both 512x512), one pass.
__global__ __launch_bounds__(256)
void nsm_init(const float* __restrict__ Mmat, float* __restrict__ P,
              float* __restrict__ I)
{
    int idx = blockIdx.x * 256 + threadIdx.x;      // < 512*512
    P[idx] = -Mmat[idx];
    int r = idx >> 9, c = idx & 511;
    I[idx] = (r == c) ? 1.0f : 0.0f;
}

extern "C" void kernel_launch(void* const* d_in, const int* in_sizes, int n_in,
                              void* d_out, int out_size, void* d_ws, size_t ws_size,
                              hipStream_t stream) {
    (void)in_sizes; (void)n_in; (void)out_size; (void)ws_size;
    const float* x  = (const float*)d_in[0];   // [1024, 4096]
    const float* W  = (const float*)d_in[1];   // [512, 1024]
    const float* Mm = (const float*)d_in[2];   // [512, 512]
    float*       out = (float*)d_out;          // [512, 4096]

    const size_t S2 = (size_t)OUT_DIM * OUT_DIM;   // 512*512
    float* Ibuf  = (float*)d_ws;
    float* P0    = Ibuf + S2;                      // P = -M
    float* Tbuf[2] = { P0 + S2, P0 + 2 * S2 };     // T ping/pong
    float* Pbuf[2] = { P0 + 3 * S2, P0 + 4 * S2 }; // P^n ping/pong
    float* A2    = P0 + 5 * S2;                    // T_101 @ W : [512,1024]
    // total ws use: 8 MB

    nsm_init<<<dim3((OUT_DIM * OUT_DIM) / 256), dim3(256), 0, stream>>>(Mm, P0, Ibuf);

    auto gemm = [&](const float* Am, int K, const float* Bm, int N,
                    const float* bias, float* Dm, int Mdim) {
        nsm_wmma_gemm<<<dim3(N / BN, Mdim / BM), dim3(256), 0, stream>>>(
            Am, K, Bm, N, bias, Dm);
    };

    // y = T_101 @ W @ x, where T_n = sum_{k=0}^{n-1} P^k, P = -M.
    // Binary powering, 101 = 0b1100101:
    //   T_2n = T_n + P^n @ T_n ; P^2n = P^n @ P^n ; T_{n+1} = I + P @ T_n.
    const int bits[6] = {1, 0, 0, 1, 0, 1};   // bits of 101 after the leading 1
    const float* T  = Ibuf;   // T_1 = I
    const float* Pn = P0;     // P^1
    int ti = 0, pi = 0;
    for (int s = 0; s < 6; ++s) {
        const bool last = (s == 5);
        float* Tn = Tbuf[ti]; ti ^= 1;
        gemm(Pn, OUT_DIM, T, OUT_DIM, T, Tn, OUT_DIM);        // T_2n
        if (!last) {
            float* Pw = Pbuf[pi]; pi ^= 1;
            gemm(Pn, OUT_DIM, Pn, OUT_DIM, nullptr, Pw, OUT_DIM);  // P^2n
            Pn = Pw;
        }
        T = Tn;
        if (bits[s]) {
            float* Tn2 = Tbuf[ti]; ti ^= 1;
            gemm(P0, OUT_DIM, T, OUT_DIM, Ibuf, Tn2, OUT_DIM);    // T_{2n+1}
            if (!last) {
                float* Pw = Pbuf[pi]; pi ^= 1;
                gemm(P0, OUT_DIM, Pn, OUT_DIM, nullptr, Pw, OUT_DIM);  // P^{2n+1}
                Pn = Pw;
            }
            T = Tn2;
        }
    }
    // n chain: 1 ->(d,+1) 3 ->(d) 6 ->(d) 12 ->(d,+1) 25 ->(d) 50 ->(d,+1) 101

    gemm(T, OUT_DIM, W, IN_DIM, nullptr, A2, OUT_DIM);   // A2 = T_101 @ W
    gemm(A2, IN_DIM, x, BATCH_N, nullptr, out, OUT_DIM); // y  = A2 @ x
}